// MTANet_7593502179851
// MI455X (gfx1250) — compile-verified
//
#include <hip/hip_runtime.h>
#include <cstdint>
#include <cstddef>

#define N_ENCS 3
#define BATCH 256
#define TLEN 168
#define DENC 32
#define UDIM 256
#define G3 768           // 3*U (gate width)
#define HOR 24
#define DEXT 16
#define K1 1280          // (U + 3U) + U  = out1 input width

typedef float v2f __attribute__((ext_vector_type(2)));
typedef float v8f __attribute__((ext_vector_type(8)));

// ---- f32 WMMA 16x16x4 helper --------------------------------------------
// A (16x4 f32): lane<16 -> {A[m][0],A[m][1]}, lane>=16 -> {A[m][2],A[m][3]}
// B (4x16 f32): lane<16 -> {B[0][n],B[1][n]}, lane>=16 -> {B[2][n],B[3][n]}
// C/D (16x16):  VGPR r, lane<16 -> (row r, col lane); lane>=16 -> (row r+8)
__device__ __forceinline__ v8f wmma4(v2f a, v2f b, v8f c) {
  return __builtin_amdgcn_wmma_f32_16x16x4_f32(false, a, false, b,
                                               (short)0, c, false, false);
}

__device__ __forceinline__ float sigmoidf_(float x) {
  return 1.0f / (1.0f + expf(-x));
}

// =========================================================================
// Encoder GRU: one workgroup per (encoder n, 16-row batch tile).
// h tile lives in LDS across all T steps. 8 waves x 6 col-tiles = 768 gates.
// =========================================================================
__global__ __launch_bounds__(256) void enc_gru_kernel(
    const float* __restrict__ x,     // [3,B,T,32]
    const float* __restrict__ W,     // [3,32,768]
    const float* __restrict__ Uw,    // [3,256,768]
    const float* __restrict__ bvec,  // [3,768]
    float* __restrict__ enc_out)     // [3,B,T,256]
{
  __shared__ float hS[16 * UDIM];    // current hidden state tile
  __shared__ float gzr[16 * 512];    // z,r gates: xg+hg+b (summed OK)
  __shared__ float gxn[16 * UDIM];   // n gate: xg+b (kept separate)
  __shared__ float ghn[16 * UDIM];   // n gate: hg

  const int n = blockIdx.x >> 4;
  const int row0 = (blockIdx.x & 15) << 4;
  const int tid = threadIdx.x;
  const int wave = tid >> 5, lane = tid & 31;
  const int half = lane >> 4, ln = lane & 15;

  const float* Wn = W + (size_t)n * DENC * G3;
  const float* Un = Uw + (size_t)n * UDIM * G3;
  const float* bn = bvec + n * G3;
  const float* xb = x + ((size_t)(n * BATCH + row0) * TLEN) * DENC;

  for (int i = tid; i < 16 * UDIM; i += 256) hS[i] = 0.0f;
  __syncthreads();

  for (int t = 0; t < TLEN; ++t) {
    const float* xt = xb + (size_t)t * DENC;
    for (int ct = 0; ct < 6; ++ct) {
      const int n0 = ((wave * 6) + ct) << 4;
      v8f ax = {};
      v8f ah = {};
      // xg contribution: K = 32
      for (int k0 = 0; k0 < DENC; k0 += 4) {
        const int k = k0 + half * 2;
        v2f a, b;
        a.x = xt[(size_t)ln * (TLEN * DENC) + k];
        a.y = xt[(size_t)ln * (TLEN * DENC) + k + 1];
        b.x = Wn[(size_t)k * G3 + n0 + ln];
        b.y = Wn[(size_t)(k + 1) * G3 + n0 + ln];
        ax = wmma4(a, b, ax);
      }
      // hg contribution: K = 256 from LDS hidden state
      for (int k0 = 0; k0 < UDIM; k0 += 4) {
        const int k = k0 + half * 2;
        v2f a, b;
        a.x = hS[ln * UDIM + k];
        a.y = hS[ln * UDIM + k + 1];
        b.x = Un[(size_t)k * G3 + n0 + ln];
        b.y = Un[(size_t)(k + 1) * G3 + n0 + ln];
        ah = wmma4(a, b, ah);
      }
      const float bb = bn[n0 + ln];
      if (n0 < 512) {
        for (int r = 0; r < 8; ++r)
          gzr[(half * 8 + r) * 512 + n0 + ln] = ax[r] + ah[r] + bb;
      } else {
        for (int r = 0; r < 8; ++r) {
          gxn[(half * 8 + r) * UDIM + (n0 - 512) + ln] = ax[r] + bb;
          ghn[(half * 8 + r) * UDIM + (n0 - 512) + ln] = ah[r];
        }
      }
    }
    __syncthreads();
    // gate combine + state update (Keras GRU: h' = (1-z)h + z*n)
    for (int i = tid; i < 16 * UDIM; i += 256) {
      const int m = i >> 8, u = i & 255;
      const float z = sigmoidf_(gzr[m * 512 + u]);
      const float r = sigmoidf_(gzr[m * 512 + 256 + u]);
      const float nn = tanhf(gxn[i] + r * ghn[i]);
      const float hnew = (1.0f - z) * hS[i] + z * nn;
      hS[i] = hnew;
      enc_out[(((size_t)(n * BATCH + row0 + m)) * TLEN + t) * UDIM + u] = hnew;
    }
    __syncthreads();
  }
}

// =========================================================================
// Generic batched GEMM, K=256, N=256:  Out[n] = A[n] @ B[n]
// grid = nMats * rowTiles; 8 waves x 2 col tiles cover 256 cols.
// Used for enc_proj (M=43008 per enc) and q = h @ att_W2 (strideAn = 0).
// =========================================================================
__global__ __launch_bounds__(256) void gemm_k256(
    const float* __restrict__ A, const float* __restrict__ Bm,
    float* __restrict__ Out, const int rowTiles,
    const size_t strideAn, const size_t strideBn, const size_t strideOn)
{
  __shared__ float aS[16 * UDIM];
  const int n = blockIdx.x / rowTiles;
  const int row0 = (blockIdx.x % rowTiles) << 4;
  const float* An = A + (size_t)n * strideAn;
  const float* Bn = Bm + (size_t)n * strideBn;
  float* On = Out + (size_t)n * strideOn;
  const int tid = threadIdx.x;
  for (int i = tid; i < 16 * UDIM; i += 256)
    aS[i] = An[(size_t)(row0 + (i >> 8)) * UDIM + (i & 255)];
  __syncthreads();
  const int wave = tid >> 5, lane = tid & 31;
  const int half = lane >> 4, ln = lane & 15;
  for (int ct = 0; ct < 2; ++ct) {
    const int n0 = ((wave * 2) + ct) << 4;
    v8f acc = {};
    for (int k0 = 0; k0 < UDIM; k0 += 4) {
      const int k = k0 + half * 2;
      v2f a, b;
      a.x = aS[ln * UDIM + k];
      a.y = aS[ln * UDIM + k + 1];
      b.x = Bn[(size_t)k * UDIM + n0 + ln];
      b.y = Bn[(size_t)(k + 1) * UDIM + n0 + ln];
      acc = wmma4(a, b, acc);
    }
    for (int r = 0; r < 8; ++r)
      On[(size_t)(row0 + half * 8 + r) * UDIM + n0 + ln] = acc[r];
  }
}

// =========================================================================
// Decoder GRU cell: x_t = [c (768) | d_t (1)];  one 16-row tile per WG.
// =========================================================================
__global__ __launch_bounds__(256) void dec_gru_kernel(
    const float* __restrict__ h_in,    // [B,256]
    const float* __restrict__ c_in,    // [B,768]
    const float* __restrict__ dec_in,  // [B,24]
    const int hstep,
    const float* __restrict__ Wd,      // [769,768]
    const float* __restrict__ Ud,      // [256,768]
    const float* __restrict__ bd,      // [768]
    float* __restrict__ h_out)         // [B,256] (may alias h_in)
{
  __shared__ float hSd[16 * UDIM];
  __shared__ float gzr[16 * 512];
  __shared__ float gxn[16 * UDIM];
  __shared__ float ghn[16 * UDIM];
  __shared__ float dS[16];

  const int row0 = blockIdx.x << 4;
  const int tid = threadIdx.x;
  for (int i = tid; i < 16 * UDIM; i += 256)
    hSd[i] = h_in[(size_t)(row0 + (i >> 8)) * UDIM + (i & 255)];
  if (tid < 16) dS[tid] = dec_in[(size_t)(row0 + tid) * HOR + hstep];
  __syncthreads();

  const int wave = tid >> 5, lane = tid & 31;
  const int half = lane >> 4, ln = lane & 15;

  for (int ct = 0; ct < 6; ++ct) {
    const int n0 = ((wave * 6) + ct) << 4;
    v8f ax = {};
    v8f ah = {};
    // x part via ctx: K = 768 (dec_W rows 0..767)
    for (int k0 = 0; k0 < G3; k0 += 4) {
      const int k = k0 + half * 2;
      v2f a, b;
      a.x = c_in[(size_t)(row0 + ln) * G3 + k];
      a.y = c_in[(size_t)(row0 + ln) * G3 + k + 1];
      b.x = Wd[(size_t)k * G3 + n0 + ln];
      b.y = Wd[(size_t)(k + 1) * G3 + n0 + ln];
      ax = wmma4(a, b, ax);
    }
    // recurrent part: K = 256
    for (int k0 = 0; k0 < UDIM; k0 += 4) {
      const int k = k0 + half * 2;
      v2f a, b;
      a.x = hSd[ln * UDIM + k];
      a.y = hSd[ln * UDIM + k + 1];
      b.x = Ud[(size_t)k * G3 + n0 + ln];
      b.y = Ud[(size_t)(k + 1) * G3 + n0 + ln];
      ah = wmma4(a, b, ah);
    }
    const float w768 = Wd[(size_t)G3 * G3 + n0 + ln];  // dec_W row 768 (d_t)
    const float bb = bd[n0 + ln];
    if (n0 < 512) {
      for (int r = 0; r < 8; ++r)
        gzr[(half * 8 + r) * 512 + n0 + ln] =
            ax[r] + dS[half * 8 + r] * w768 + bb + ah[r];
    } else {
      for (int r = 0; r < 8; ++r) {
        gxn[(half * 8 + r) * UDIM + (n0 - 512) + ln] =
            ax[r] + dS[half * 8 + r] * w768 + bb;
        ghn[(half * 8 + r) * UDIM + (n0 - 512) + ln] = ah[r];
      }
    }
  }
  __syncthreads();
  for (int i = tid; i < 16 * UDIM; i += 256) {
    const int m = i >> 8, u = i & 255;
    const float z = sigmoidf_(gzr[m * 512 + u]);
    const float r = sigmoidf_(gzr[m * 512 + 256 + u]);
    const float nn = tanhf(gxn[i] + r * ghn[i]);
    h_out[(size_t)(row0 + m) * UDIM + u] = (1.0f - z) * hSd[i] + z * nn;
  }
}

// =========================================================================
// Bahdanau attention for one (encoder n, batch b): score -> softmax -> ctx
// =========================================================================
__global__ __launch_bounds__(256) void attn_kernel(
    const float* __restrict__ enc_out,   // [3,B,T,256]
    const float* __restrict__ enc_proj,  // [3,B,T,256]
    const float* __restrict__ q,         // [3,B,256]
    const float* __restrict__ att_v,     // [3,256]
    float* __restrict__ c_ctx)           // [B,768]
{
  const int n = blockIdx.x >> 8;
  const int b = blockIdx.x & 255;
  __shared__ float qS[UDIM];
  __shared__ float avS[UDIM];
  __shared__ float sS[TLEN];
  const int tid = threadIdx.x;
  qS[tid] = q[((size_t)n * BATCH + b) * UDIM + tid];
  avS[tid] = att_v[n * UDIM + tid];
  __syncthreads();
  const float* proj = enc_proj + ((size_t)(n * BATCH + b)) * TLEN * UDIM;
  for (int t = tid; t < TLEN; t += 256) {
    float s = 0.0f;
    const float* pr = proj + (size_t)t * UDIM;
    for (int v = 0; v < UDIM; ++v) s += tanhf(pr[v] + qS[v]) * avS[v];
    sS[t] = s;
  }
  __syncthreads();
  if (tid == 0) {  // serial softmax over T=168: negligible work
    float m = sS[0];
    for (int t = 1; t < TLEN; ++t) m = fmaxf(m, sS[t]);
    float sum = 0.0f;
    for (int t = 0; t < TLEN; ++t) { float e = expf(sS[t] - m); sS[t] = e; sum += e; }
    const float inv = 1.0f / sum;
    for (int t = 0; t < TLEN; ++t) sS[t] *= inv;
  }
  __syncthreads();
  const float* eo = enc_out + ((size_t)(n * BATCH + b)) * TLEN * UDIM;
  float acc = 0.0f;
  for (int t = 0; t < TLEN; ++t) acc += sS[t] * eo[(size_t)t * UDIM + tid];
  c_ctx[(size_t)b * G3 + n * UDIM + tid] = acc;
}

// =========================================================================
// Output head
// =========================================================================
__global__ __launch_bounds__(256) void assemble_kernel(
    const float* __restrict__ h_dec, const float* __restrict__ c_ctx,
    const float* __restrict__ ext_feat, const float* __restrict__ ext_W,
    const float* __restrict__ ext_b, float* __restrict__ X)
{
  __shared__ float ef[DEXT];
  const int row = blockIdx.x;  // b*HOR + h
  const int b = row / HOR;
  const int tid = threadIdx.x;
  if (tid < DEXT) ef[tid] = ext_feat[(size_t)row * DEXT + tid];
  __syncthreads();
  float* Xr = X + (size_t)row * K1;
  Xr[tid] = h_dec[(size_t)b * UDIM + tid];
  Xr[256 + tid] = c_ctx[(size_t)b * G3 + tid];
  Xr[512 + tid] = c_ctx[(size_t)b * G3 + 256 + tid];
  Xr[768 + tid] = c_ctx[(size_t)b * G3 + 512 + tid];
  float s = ext_b[tid];
  for (int k = 0; k < DEXT; ++k) s += ef[k] * ext_W[k * UDIM + tid];
  Xr[1024 + tid] = fmaxf(s, 0.0f);
}

__global__ __launch_bounds__(256) void out1_kernel(
    const float* __restrict__ X, const float* __restrict__ Wm,
    const float* __restrict__ bias, float* __restrict__ Y)
{
  __shared__ float aS[16 * K1];
  const int row0 = blockIdx.x << 4;
  const int tid = threadIdx.x;
  for (int i = tid; i < 16 * K1; i += 256)
    aS[i] = X[(size_t)(row0 + i / K1) * K1 + (i % K1)];
  __syncthreads();
  const int wave = tid >> 5, lane = tid & 31;
  const int half = lane >> 4, ln = lane & 15;
  for (int ct = 0; ct < 2; ++ct) {
    const int n0 = ((wave * 2) + ct) << 4;
    v8f acc = {};
    for (int k0 = 0; k0 < K1; k0 += 4) {
      const int k = k0 + half * 2;
      v2f a, b;
      a.x = aS[ln * K1 + k];
      a.y = aS[ln * K1 + k + 1];
      b.x = Wm[(size_t)k * UDIM + n0 + ln];
      b.y = Wm[(size_t)(k + 1) * UDIM + n0 + ln];
      acc = wmma4(a, b, acc);
    }
    const float bb = bias[n0 + ln];
    for (int r = 0; r < 8; ++r)
      Y[(size_t)(row0 + half * 8 + r) * UDIM + n0 + ln] =
          fmaxf(acc[r] + bb, 0.0f);
  }
}

__global__ __launch_bounds__(256) void out2_kernel(
    const float* __restrict__ X1, const float* __restrict__ w2,
    const float* __restrict__ b2, float* __restrict__ out)
{
  __shared__ float wS[UDIM];
  const int tid = threadIdx.x;
  wS[tid] = w2[tid];
  __syncthreads();
  const int i = blockIdx.x * 256 + tid;  // one of B*HOR rows per thread
  const float* row = X1 + (size_t)i * UDIM;
  float acc = b2[0];
  for (int k = 0; k < UDIM; ++k) acc += row[k] * wS[k];
  out[i] = acc;
}

__global__ __launch_bounds__(256) void zero_kernel(float* p, const int nElem) {
  const int i = blockIdx.x * 256 + threadIdx.x;
  if (i < nElem) p[i] = 0.0f;
}

// =========================================================================
extern "C" void kernel_launch(void* const* d_in, const int* in_sizes, int n_in,
                              void* d_out, int out_size, void* d_ws,
                              size_t ws_size, hipStream_t stream) {
  (void)in_sizes; (void)n_in; (void)out_size; (void)ws_size;
  const float* enc_inputs = (const float*)d_in[0];
  const float* dec_input  = (const float*)d_in[1];
  const float* ext_feat   = (const float*)d_in[2];
  const float* enc_W      = (const float*)d_in[3];
  const float* enc_U      = (const float*)d_in[4];
  const float* enc_b      = (const float*)d_in[5];
  const float* att_W1     = (const float*)d_in[6];
  const float* att_W2     = (const float*)d_in[7];
  const float* att_v      = (const float*)d_in[8];
  const float* dec_W      = (const float*)d_in[9];
  const float* dec_U      = (const float*)d_in[10];
  const float* dec_b      = (const float*)d_in[11];
  const float* ext_W      = (const float*)d_in[12];
  const float* ext_b      = (const float*)d_in[13];
  const float* out1_W     = (const float*)d_in[14];
  const float* out1_b     = (const float*)d_in[15];
  const float* out2_W     = (const float*)d_in[16];
  const float* out2_b     = (const float*)d_in[17];

  float* ws = (float*)d_ws;
  const size_t ENCOUT = (size_t)N_ENCS * BATCH * TLEN * UDIM;  // 33,030,144
  float* enc_out  = ws;
  float* enc_proj = ws + ENCOUT;
  float* h_dec    = ws + 2 * ENCOUT;
  float* c_ctx    = h_dec + (size_t)BATCH * UDIM;
  float* q_buf    = c_ctx + (size_t)BATCH * G3;
  float* x1in     = q_buf + (size_t)N_ENCS * BATCH * UDIM;
  float* x1       = x1in + (size_t)BATCH * HOR * K1;

  // 1) encoders (recurrent, h kept in LDS per batch tile)
  enc_gru_kernel<<<N_ENCS * (BATCH / 16), 256, 0, stream>>>(
      enc_inputs, enc_W, enc_U, enc_b, enc_out);

  // 2) Bahdanau keys: enc_proj = enc_out @ att_W1
  const int MROWS = BATCH * TLEN;  // 43008 rows per encoder
  gemm_k256<<<N_ENCS * (MROWS / 16), 256, 0, stream>>>(
      enc_out, att_W1, enc_proj, MROWS / 16,
      (size_t)MROWS * UDIM, (size_t)UDIM * UDIM, (size_t)MROWS * UDIM);

  // 3) decoder init: h0 = 0, c0 = 0 (contiguous in ws)
  zero_kernel<<<(BATCH * UDIM + BATCH * G3) / 256, 256, 0, stream>>>(
      h_dec, BATCH * UDIM + BATCH * G3);

  // 4) decoder scan
  for (int hs = 0; hs < HOR; ++hs) {
    dec_gru_kernel<<<BATCH / 16, 256, 0, stream>>>(
        h_dec, c_ctx, dec_input, hs, dec_W, dec_U, dec_b, h_dec);
    gemm_k256<<<N_ENCS * (BATCH / 16), 256, 0, stream>>>(
        h_dec, att_W2, q_buf, BATCH / 16,
        (size_t)0, (size_t)UDIM * UDIM, (size_t)BATCH * UDIM);
    attn_kernel<<<N_ENCS * BATCH, 256, 0, stream>>>(
        enc_out, enc_proj, q_buf, att_v, c_ctx);
  }

  // 5) output head
  assemble_kernel<<<BATCH * HOR, 256, 0, stream>>>(
      h_dec, c_ctx, ext_feat, ext_W, ext_b, x1in);
  out1_kernel<<<(BATCH * HOR) / 16, 256, 0, stream>>>(x1in, out1_W, out1_b, x1);
  out2_kernel<<<(BATCH * HOR) / 256, 256, 0, stream>>>(
      x1, out2_W, out2_b, (float*)d_out);
}